// VariationalGraphExtractor_68264210203182
// MI455X (gfx1250) — compile-verified
//
#include <hip/hip_runtime.h>

typedef __attribute__((ext_vector_type(2))) float v2f;
typedef __attribute__((ext_vector_type(8))) float v8f;

#define B_   16
#define S_   4096
#define D_   1024
#define NS_  64
#define THREADS 256
#define COLS_PER_WG ((THREADS / 32) * 16)   // 8 waves * 16 cols = 128

#define WMMA_ACC(bv, c) \
    __builtin_amdgcn_wmma_f32_16x16x4_f32(false, a, false, (bv), (short)0, (c), false, false)

// One workgroup per (batch, segment). Row-sum over the segment done on the
// matrix pipe: D = ones(16x4) * B(4x16) + C accumulates column sums in fp32.
__global__ __launch_bounds__(THREADS)
void seg_mean_wmma_kernel(const int* __restrict__ sent_ind,
                          const float* __restrict__ x,
                          float* __restrict__ out)
{
    const int wg  = blockIdx.x;        // 0 .. B*NS-1
    const int b   = wg / NS_;
    const int seg = wg % NS_;

    const int* __restrict__ si = sent_ind + (size_t)b * S_;

    // lower_bound(si, seg): first row of this segment (sent_ind is sorted)
    int lo;
    {
        int l = 0, r = S_;
        while (l < r) { int m = (l + r) >> 1; if (si[m] < seg) l = m + 1; else r = m; }
        lo = l;
    }
    // lower_bound(si, seg+1): one past last row
    int hi;
    {
        int l = lo, r = S_;
        while (l < r) { int m = (l + r) >> 1; if (si[m] < seg + 1) l = m + 1; else r = m; }
        hi = l;
    }
    const int count     = hi - lo;
    const int fullSteps = count >> 2;    // whole 4-row WMMA steps, no bounds checks
    const int rem       = count & 3;     // ragged tail rows (0 for the actual data)

    const int tid     = (int)threadIdx.x;
    const int lane    = tid & 31;
    const int wave    = tid >> 5;
    const int halfSel = lane >> 4;       // which K-pair this lane feeds
    const int nIdx    = lane & 15;       // column slot (lanes n and n+16 share it)

    const float* __restrict__ xb = x + (size_t)b * S_ * D_;
    float* __restrict__ ob = out + (size_t)b * (NS_ + 1) * D_;

    v2f a; a.x = 1.0f; a.y = 1.0f;       // all-ones A matrix (constant)

    for (int colBase = wave * 16; colBase < D_; colBase += COLS_PER_WG) {
        const int col = colBase + nIdx;

        v8f c = {0.f, 0.f, 0.f, 0.f, 0.f, 0.f, 0.f, 0.f};

        // Base pointer for this lane: row (lo + 2*halfSel), column col.
        // All row offsets below are compile-time constants -> 24-bit IOFFSET,
        // so the unrolled body is 8 plain global_load_b32 + 4 WMMAs.
        const float* p = xb + (size_t)(lo + halfSel * 2) * D_ + col;

        int t = 0;
        for (; t + 4 <= fullSteps; t += 4) {   // 16 rows per iteration
            v2f b0, b1, b2, b3;
            b0.x = p[ 0 * D_]; b0.y = p[ 1 * D_];
            b1.x = p[ 4 * D_]; b1.y = p[ 5 * D_];
            b2.x = p[ 8 * D_]; b2.y = p[ 9 * D_];
            b3.x = p[12 * D_]; b3.y = p[13 * D_];
            c = WMMA_ACC(b0, c);
            c = WMMA_ACC(b1, c);
            c = WMMA_ACC(b2, c);
            c = WMMA_ACC(b3, c);
            p += 16 * D_;
        }
        for (; t < fullSteps; ++t) {           // leftover whole steps (0..3)
            v2f bv;
            bv.x = p[0 * D_];
            bv.y = p[1 * D_];
            c = WMMA_ACC(bv, c);
            p += 4 * D_;
        }
        if (rem) {                             // ragged tail, at most once
            const int r0 = lo + fullSteps * 4 + halfSel * 2;
            const int r1 = r0 + 1;
            v2f bv;
            bv.x = (r0 < hi) ? xb[(size_t)r0 * D_ + col] : 0.0f;
            bv.y = (r1 < hi) ? xb[(size_t)r1 * D_ + col] : 0.0f;
            c = WMMA_ACC(bv, c);
        }

        const float sum = c[0];                // M=0 (or M=8) row: total column sum

        if (lane < 16) {                       // lanes n and n+16 are duplicates
            if (seg == 0) {
                const float first = xb[col];   // global row 0 lives in segment 0
                ob[col] = first;                                           // slot 0
                ob[(size_t)D_ + col] = (sum - first) / (float)(count - 1); // slot 1
            } else {
                ob[(size_t)(seg + 1) * D_ + col] = sum / (float)count;     // slots 2..NS
            }
        }
    }
}

extern "C" void kernel_launch(void* const* d_in, const int* in_sizes, int n_in,
                              void* d_out, int out_size, void* d_ws, size_t ws_size,
                              hipStream_t stream)
{
    (void)in_sizes; (void)n_in; (void)out_size; (void)d_ws; (void)ws_size;

    const int*   sent_ind = (const int*)d_in[0];
    const float* x        = (const float*)d_in[1];
    // d_in[2] = num_sent (scalar on device); dims fixed by the reference problem.
    float* out = (float*)d_out;

    dim3 grid(B_ * NS_);
    dim3 block(THREADS);
    seg_mean_wmma_kernel<<<grid, block, 0, stream>>>(sent_ind, x, out);
}